// HybridGNN_77936476553793
// MI455X (gfx1250) — compile-verified
//
#include <hip/hip_runtime.h>
#include <hip/hip_bf16.h>

typedef __attribute__((ext_vector_type(2))) float v2f;
typedef __attribute__((ext_vector_type(8))) float v8f;

#define DEMB 64

// ---------------------------------------------------------------- utilities
__global__ void k_fill(float* __restrict__ p, float v, size_t n) {
  size_t i = (size_t)blockIdx.x * blockDim.x + threadIdx.x;
  size_t stride = (size_t)gridDim.x * blockDim.x;
  for (; i < n; i += stride) p[i] = v;
}

__device__ inline void atomicMaxFloat(float* addr, float v) {
  // monotonic trick: works for mixed signs with -inf init
  if (v >= 0.0f) atomicMax((int*)addr, __float_as_int(v));
  else           atomicMin((unsigned int*)addr, __float_as_uint(v));
}

// --------------------------------------------- WMMA f32 16x16x4 GEMM + LDS
// C[M, NT*16] = act(A[M,K] @ B[K, NT*16] + bias).  Whole B staged into LDS
// via async-to-LDS; 8 waves per block, each wave computes a 16-row strip.
// Inner K-step: prefetch next A frag (global), batch all NT B-frag LDS
// loads, then NT WMMAs -> one ds wait per step, NT independent acc chains.
// ACT: 0 = none, 1 = relu
template <int NT, int ACT>
__global__ void k_gemm_lds(const float* __restrict__ A, const float* __restrict__ Bm,
                           const float* __restrict__ bias, float* __restrict__ C,
                           int M, int K) {
  constexpr int Nc = NT * 16;
  extern __shared__ float Bs[];
  const int tid = threadIdx.x;
  const int nB  = K * Nc;

#if defined(__gfx1250__) && __has_builtin(__builtin_amdgcn_global_load_async_to_lds_b32) && \
    __has_builtin(__builtin_amdgcn_s_wait_asynccnt)
  for (int i = tid; i < nB; i += (int)blockDim.x) {
    __builtin_amdgcn_global_load_async_to_lds_b32(
        (__attribute__((address_space(1))) int*)(Bm + i),
        (__attribute__((address_space(3))) int*)(Bs + i), 0, 0);
  }
  __builtin_amdgcn_s_wait_asynccnt(0);
#else
  for (int i = tid; i < nB; i += (int)blockDim.x) Bs[i] = Bm[i];
#endif
  __syncthreads();

  const int wave = tid >> 5;                // 8 waves -> 128 rows per block
  const int lane = tid & 31;
  const int col  = lane & 15;
  const int koff = (lane >> 4) << 1;        // K base for this half-wave
  const int row  = blockIdx.x * 128 + wave * 16 + col;
  const int row_c = row < M ? row : M - 1;  // clamp: EXEC stays all-ones for WMMA
  const float* Arow = A + (size_t)row_c * K;

  v8f acc[NT] = {};

  v2f a;                                    // software-pipelined A fragment
  a.x = Arow[koff];
  a.y = Arow[koff + 1];

  for (int k = 0; k < K; k += 4) {
    const int kb = k + koff;
    v2f a_next = a;
    if (k + 4 < K) {                        // prefetch next A frag (global)
      a_next.x = Arow[kb + 4];
      a_next.y = Arow[kb + 5];
    }
    v2f bf[NT];
#pragma unroll
    for (int t = 0; t < NT; ++t) {          // batch all LDS loads first
      bf[t].x = Bs[kb * Nc + t * 16 + col];
      bf[t].y = Bs[(kb + 1) * Nc + t * 16 + col];
    }
#pragma unroll
    for (int t = 0; t < NT; ++t) {
      acc[t] = __builtin_amdgcn_wmma_f32_16x16x4_f32(false, a, false, bf[t],
                                                     (short)0, acc[t], false, false);
    }
    a = a_next;
  }

  const int rbase = blockIdx.x * 128 + wave * 16 + ((lane >> 4) << 3);
#pragma unroll
  for (int t = 0; t < NT; ++t) {
    const int cg = t * 16 + col;
    const float bv = bias ? bias[cg] : 0.0f;
#pragma unroll
    for (int i = 0; i < 8; ++i) {
      const int r = rbase + i;
      if (r < M) {
        float v = acc[t][i] + bv;
        if (ACT == 1) v = v > 0.0f ? v : 0.0f;
        C[(size_t)r * Nc + cg] = v;
      }
    }
  }
}

// -------------------------------------------- node feature assembly  x[N,2D]
__global__ void k_build_x(const int* __restrict__ uid, const int* __restrict__ iid,
                          const float* __restrict__ uembw, const float* __restrict__ iembw,
                          const float* __restrict__ cemb, float* __restrict__ x,
                          int Bn, int NU) {
  int t = blockIdx.x * blockDim.x + threadIdx.x;
  if (t >= Bn * DEMB) return;
  int b = t >> 6, c = t & 63;
  int u = uid[b], it = iid[b];
  x[(size_t)u * 128 + c]              = uembw[(size_t)u * DEMB + c];
  x[(size_t)(NU + it) * 128 + c]      = iembw[(size_t)it * DEMB + c];
  x[(size_t)(NU + it) * 128 + 64 + c] = cemb[(size_t)b * DEMB + c];
}

// ------------------------------------- attention logits  als/ald [N,H]
__global__ void k_attn(const float* __restrict__ xw, const float* __restrict__ asrc,
                       const float* __restrict__ adst, float* __restrict__ als,
                       float* __restrict__ ald, int Nn, int H) {
  int idx = blockIdx.x * blockDim.x + threadIdx.x;
  if (idx >= Nn * H) return;
  int n = idx / H, h = idx - n * H;
  const float* row = xw + ((size_t)n * H + h) * DEMB;
  const float* as  = asrc + h * DEMB;
  const float* ad  = adst + h * DEMB;
  float ss = 0.0f, sd = 0.0f;
  for (int c = 0; c < DEMB; ++c) { float v = row[c]; ss += v * as[c]; sd += v * ad[c]; }
  als[idx] = ss;
  ald[idx] = sd;
}

__device__ inline void edge_sd(const int* __restrict__ ei, long e, long E, int& s, int& d) {
  if (e < E) { s = ei[e]; d = ei[E + e]; }
  else       { s = d = (int)(e - E); }     // self loops
}

__device__ inline float leaky(float t) { return t > 0.0f ? t : 0.2f * t; }

// ---------------------------------------------------- edge pass 1: seg-max
__global__ void k_edge_max(const int* __restrict__ ei, const float* __restrict__ als,
                           const float* __restrict__ ald, float* __restrict__ m,
                           long E, long Et, int H) {
  long idx = (long)blockIdx.x * blockDim.x + threadIdx.x;
  if (idx >= Et * H) return;
  long e = idx / H; int h = (int)(idx - e * H);
  int s, d; edge_sd(ei, e, E, s, d);
  float t = leaky(als[(size_t)s * H + h] + ald[(size_t)d * H + h]);
  atomicMaxFloat(&m[(size_t)d * H + h], t);
}

// ------------------------------------------------- edge pass 2: exp-sum
__global__ void k_edge_expsum(const int* __restrict__ ei, const float* __restrict__ als,
                              const float* __restrict__ ald, const float* __restrict__ m,
                              float* __restrict__ den, long E, long Et, int H) {
  long idx = (long)blockIdx.x * blockDim.x + threadIdx.x;
  if (idx >= Et * H) return;
  long e = idx / H; int h = (int)(idx - e * H);
  int s, d; edge_sd(ei, e, E, s, d);
  float t = leaky(als[(size_t)s * H + h] + ald[(size_t)d * H + h]);
  atomicAdd(&den[(size_t)d * H + h], __expf(t - m[(size_t)d * H + h]));
}

// --------------------------------------- edge pass 3: weighted aggregation
// 16 consecutive lanes cover one (edge, head) row (float4 each) so the
// xw[src] gather and the out[dst] atomic scatter are fully coalesced.
__global__ void k_edge_agg(const int* __restrict__ ei, const float* __restrict__ als,
                           const float* __restrict__ ald, const float* __restrict__ m,
                           const float* __restrict__ den, const float* __restrict__ xw,
                           float* __restrict__ out, long E, long Et, int H) {
  long idx = (long)blockIdx.x * blockDim.x + threadIdx.x;
  if (idx >= Et * H * 16) return;
  const int c4 = (int)(idx & 15);
  long eh = idx >> 4;
  long e = eh / H; int h = (int)(eh - e * H);
  int s, d; edge_sd(ei, e, E, s, d);
  float t = leaky(als[(size_t)s * H + h] + ald[(size_t)d * H + h]);
  float alpha = __expf(t - m[(size_t)d * H + h]) / (den[(size_t)d * H + h] + 1e-16f);
  const float4 v = ((const float4*)(xw + ((size_t)s * H + h) * DEMB))[c4];
  float* drow = out + ((size_t)d * H + h) * DEMB + 4 * c4;
  atomicAdd(drow + 0, v.x * alpha);
  atomicAdd(drow + 1, v.y * alpha);
  atomicAdd(drow + 2, v.z * alpha);
  atomicAdd(drow + 3, v.w * alpha);
}

// ----------------------------------------- bias (+ optional ELU), in place
__global__ void k_bias_act(float* __restrict__ p, const float* __restrict__ bias,
                           size_t n, int cols, int act) {
  size_t i = (size_t)blockIdx.x * blockDim.x + threadIdx.x;
  size_t stride = (size_t)gridDim.x * blockDim.x;
  for (; i < n; i += stride) {
    float v = p[i] + bias[i % cols];
    if (act == 1) v = v > 0.0f ? v : (__expf(v) - 1.0f);  // ELU
    p[i] = v;
  }
}

// -------------------------------------------------- combined [B, 3D] build
__global__ void k_combine(const int* __restrict__ uid, const int* __restrict__ iid,
                          const float* __restrict__ uembw, const float* __restrict__ x2,
                          float* __restrict__ comb, int Bn, int NU) {
  int t = blockIdx.x * blockDim.x + threadIdx.x;
  if (t >= Bn * DEMB) return;
  int b = t >> 6, c = t & 63;
  int u = uid[b], it = iid[b];
  comb[(size_t)b * 192 + c]       = uembw[(size_t)u * DEMB + c];
  comb[(size_t)b * 192 + 64 + c]  = x2[(size_t)(NU + it) * DEMB + c];
  comb[(size_t)b * 192 + 128 + c] = x2[(size_t)u * DEMB + c];
}

// --------------------------------------------------------- final projection
__global__ void k_final(const float* __restrict__ h, const float* __restrict__ pw2,
                        const float* __restrict__ pb2, float* __restrict__ out, int Bn) {
  int b = blockIdx.x * blockDim.x + threadIdx.x;
  if (b >= Bn) return;
  float s = pb2[0];
  for (int c = 0; c < DEMB; ++c) s += h[(size_t)b * DEMB + c] * pw2[c];
  out[b] = s;
}

// ===========================================================================
extern "C" void kernel_launch(void* const* d_in, const int* in_sizes, int n_in,
                              void* d_out, int out_size, void* d_ws, size_t ws_size,
                              hipStream_t stream) {
  const int*   user_ids = (const int*)d_in[0];
  const int*   item_ids = (const int*)d_in[1];
  const float* content  = (const float*)d_in[2];
  const int*   edge     = (const int*)d_in[3];
  const float* uembw    = (const float*)d_in[4];
  const float* iembw    = (const float*)d_in[5];
  const float* cpw      = (const float*)d_in[6];
  const float* cpb      = (const float*)d_in[7];
  const float* w1       = (const float*)d_in[8];
  const float* asrc1    = (const float*)d_in[9];
  const float* adst1    = (const float*)d_in[10];
  const float* b1       = (const float*)d_in[11];
  const float* w2       = (const float*)d_in[12];
  const float* asrc2    = (const float*)d_in[13];
  const float* adst2    = (const float*)d_in[14];
  const float* b2       = (const float*)d_in[15];
  const float* pw1      = (const float*)d_in[16];
  const float* pb1      = (const float*)d_in[17];
  const float* pw2      = (const float*)d_in[18];
  const float* pb2      = (const float*)d_in[19];
  float* out = (float*)d_out;

  const int  Bn = in_sizes[0];
  const long E  = (long)in_sizes[3] / 2;
  const int  NU = in_sizes[4] / DEMB;
  const int  NI = in_sizes[5] / DEMB;
  const int  Nn = NU + NI;
  const int  CF = in_sizes[2] / Bn;
  const long Et = E + Nn;

  // ---- workspace layout (floats) ----
  float* X    = (float*)d_ws;                 // [N,128]  x, then GAT1 out
  float* XW   = X    + (size_t)Nn * 128;      // [N,128]  xw1, then xw2
  float* ALS  = XW   + (size_t)Nn * 128;      // [N,2]
  float* ALD  = ALS  + (size_t)Nn * 2;        // [N,2]
  float* MM   = ALD  + (size_t)Nn * 2;        // [N,2]
  float* DEN  = MM   + (size_t)Nn * 2;        // [N,2]
  float* OUT2 = DEN  + (size_t)Nn * 2;        // [N,64]
  float* CEMB = OUT2 + (size_t)Nn * 64;       // [B,64]
  float* COMB = CEMB + (size_t)Bn * 64;       // [B,192]
  float* HH   = COMB + (size_t)Bn * 192;      // [B,64]

  auto blocks = [](long n, int t) { return (int)((n + t - 1) / t); };
  const float NEGINF = -__builtin_inff();

  // 1) x = 0 ; c_emb = content @ cp_w + cp_b ; scatter rows
  k_fill<<<blocks((long)Nn * 128, 256), 256, 0, stream>>>(X, 0.0f, (size_t)Nn * 128);
  k_gemm_lds<4, 0><<<blocks(Bn, 128), 256, (size_t)CF * DEMB * 4, stream>>>(content, cpw, cpb, CEMB, Bn, CF);
  k_build_x<<<blocks((long)Bn * DEMB, 256), 256, 0, stream>>>(user_ids, item_ids, uembw, iembw, CEMB, X, Bn, NU);

  // 2) GAT layer 1 (H=2, C=64): xw1 = x @ w1
  k_gemm_lds<8, 0><<<blocks(Nn, 128), 256, (size_t)128 * 128 * 4, stream>>>(X, w1, nullptr, XW, Nn, 128);
  k_attn<<<blocks((long)Nn * 2, 256), 256, 0, stream>>>(XW, asrc1, adst1, ALS, ALD, Nn, 2);
  k_fill<<<blocks((long)Nn * 2, 256), 256, 0, stream>>>(MM, NEGINF, (size_t)Nn * 2);
  k_fill<<<blocks((long)Nn * 2, 256), 256, 0, stream>>>(DEN, 0.0f, (size_t)Nn * 2);
  k_fill<<<blocks((long)Nn * 128, 256), 256, 0, stream>>>(X, 0.0f, (size_t)Nn * 128);  // reuse as out1
  k_edge_max<<<blocks(Et * 2, 256), 256, 0, stream>>>(edge, ALS, ALD, MM, E, Et, 2);
  k_edge_expsum<<<blocks(Et * 2, 256), 256, 0, stream>>>(edge, ALS, ALD, MM, DEN, E, Et, 2);
  k_edge_agg<<<blocks(Et * 2 * 16, 256), 256, 0, stream>>>(edge, ALS, ALD, MM, DEN, XW, X, E, Et, 2);
  k_bias_act<<<blocks((long)Nn * 128, 256), 256, 0, stream>>>(X, b1, (size_t)Nn * 128, 128, 1); // +b1, ELU

  // 3) GAT layer 2 (H=1, C=64): xw2 = x @ w2
  k_gemm_lds<4, 0><<<blocks(Nn, 128), 256, (size_t)128 * DEMB * 4, stream>>>(X, w2, nullptr, XW, Nn, 128);
  k_attn<<<blocks((long)Nn, 256), 256, 0, stream>>>(XW, asrc2, adst2, ALS, ALD, Nn, 1);
  k_fill<<<blocks((long)Nn, 256), 256, 0, stream>>>(MM, NEGINF, (size_t)Nn);
  k_fill<<<blocks((long)Nn, 256), 256, 0, stream>>>(DEN, 0.0f, (size_t)Nn);
  k_fill<<<blocks((long)Nn * 64, 256), 256, 0, stream>>>(OUT2, 0.0f, (size_t)Nn * 64);
  k_edge_max<<<blocks(Et, 256), 256, 0, stream>>>(edge, ALS, ALD, MM, E, Et, 1);
  k_edge_expsum<<<blocks(Et, 256), 256, 0, stream>>>(edge, ALS, ALD, MM, DEN, E, Et, 1);
  k_edge_agg<<<blocks(Et * 16, 256), 256, 0, stream>>>(edge, ALS, ALD, MM, DEN, XW, OUT2, E, Et, 1);
  k_bias_act<<<blocks((long)Nn * 64, 256), 256, 0, stream>>>(OUT2, b2, (size_t)Nn * 64, 64, 0); // +b2

  // 4) predict head
  k_combine<<<blocks((long)Bn * DEMB, 256), 256, 0, stream>>>(user_ids, item_ids, uembw, OUT2, COMB, Bn, NU);
  k_gemm_lds<4, 1><<<blocks(Bn, 128), 256, (size_t)192 * DEMB * 4, stream>>>(COMB, pw1, pb1, HH, Bn, 192); // ReLU
  k_final<<<blocks((long)Bn, 256), 256, 0, stream>>>(HH, pw2, pb2, out, Bn);
}